// SNN_55937654063543
// MI455X (gfx1250) — compile-verified
//
#include <hip/hip_runtime.h>
#include <math.h>

// ---------------- problem constants (match reference) ----------------
constexpr int kN0 = 131072, kN1 = 262144, kN2 = 131072;
constexpr int kDEG = 12;
constexpr int kF1 = 64, kF2 = 32, kF3 = 32;
constexpr int kH = 32;
constexpr int kB = 256;
constexpr int kOUT = 10;
constexpr float kSLOPE = 0.01f;

typedef float v2f __attribute__((ext_vector_type(2)));
typedef float v8f __attribute__((ext_vector_type(8)));

// ---------------- zero fill ----------------
__global__ void fill0(float* __restrict__ p, long long n) {
    long long i = (long long)blockIdx.x * blockDim.x + threadIdx.x;
    if (i < n) p[i] = 0.0f;
}

// ---------------- SpMM: y[row] += val * act(x[col]) ----------------
// One wave handles 32 edges: lane e loads edge metadata, then broadcast via
// shfl; lane = feature index -> x[col*F + lane] is one coalesced 128B line.
template <int F, int LEAKY>
__global__ void spmm_kernel(const int* __restrict__ row,
                            const int* __restrict__ col,
                            const float* __restrict__ val,
                            const float* __restrict__ x,
                            float* __restrict__ y,
                            long long nnz) {
    const int lane = threadIdx.x & 31;
    const long long base =
        ((long long)blockIdx.x * (blockDim.x >> 5) + (threadIdx.x >> 5)) * 32;
    if (base >= nnz) return;

    int r = row[base + lane];
    int c = col[base + lane];
    float v = val[base + lane];

    #pragma unroll 4
    for (int e = 0; e < 32; ++e) {
        int   re = __shfl(r, e);
        int   ce = __shfl(c, e);
        float ve = __shfl(v, e);
        if (e + 1 < 32) {  // hide gather latency: global_prefetch_b8
            int cn = __shfl(c, e + 1);
            __builtin_prefetch(&x[(size_t)cn * F + lane], 0, 1);
        }
        #pragma unroll
        for (int f = lane; f < F; f += 32) {
            float xv = x[(size_t)ce * F + f];
            if (LEAKY) xv = (xv >= 0.0f) ? xv : kSLOPE * xv;
            atomicAdd(&y[(size_t)re * F + f], ve * xv);
        }
    }
}

// ---------------- dense GEMM via V_WMMA_F32_16X16X4_F32 ----------------
// Hout[n x 32] = Z[n x FIN] @ W[FIN x 32] + bias.  One wave per 16x16 tile,
// ntile in {0,1} covers the 32 output columns.  FIN in {32,64} -> 8/16 WMMAs.
template <int FIN>
__global__ void gemm_wmma(const float* __restrict__ Z,
                          const float* __restrict__ W,
                          const float* __restrict__ bias,
                          float* __restrict__ Hout) {
    const int lane  = threadIdx.x & 31;
    const int tile  = blockIdx.x * (blockDim.x >> 5) + (threadIdx.x >> 5);
    const int mtile = tile >> 1;
    const int ntile = tile & 1;
    const int m0    = mtile * 16;
    const int lm    = lane & 15;          // M (for A) / N (for B,C)
    const int kgrp  = (lane >> 4) * 2;    // K sub-offset for this half-wave
    const int ncol  = ntile * 16 + lm;

    v8f c;
    {
        const float b0 = bias[ncol];      // bias depends on column only
        #pragma unroll
        for (int r = 0; r < 8; ++r) c[r] = b0;
    }

    #pragma unroll
    for (int k0 = 0; k0 < FIN; k0 += 4) {
        const float* arow = Z + (size_t)(m0 + lm) * FIN + (k0 + kgrp);
        v2f a; a.x = arow[0]; a.y = arow[1];
        const float* brow = W + (size_t)(k0 + kgrp) * kH + ncol;
        v2f b; b.x = brow[0]; b.y = brow[kH];
        c = __builtin_amdgcn_wmma_f32_16x16x4_f32(
                /*neg_a=*/false, a, /*neg_b=*/false, b,
                /*c_mod=*/(short)0, c, /*reuse_a=*/false, /*reuse_b=*/false);
    }

    // C/D layout: VGPR r -> row r (lanes 0-15) / row r+8 (lanes 16-31)
    const int rbase = m0 + ((lane >> 4) ? 8 : 0);
    #pragma unroll
    for (int r = 0; r < 8; ++r)
        Hout[(size_t)(rbase + r) * kH + ncol] = c[r];
}

// ---------------- mean pooling ----------------
__global__ void pool_sum(const float* __restrict__ h,
                         const int* __restrict__ batch,
                         float* __restrict__ s) {
    long long idx = (long long)blockIdx.x * blockDim.x + threadIdx.x;
    int node = (int)(idx >> 5);
    int f = (int)(idx & 31);
    int g = batch[node];
    atomicAdd(&s[(size_t)g * kH + f], h[idx]);
}

__global__ void pool_count(const int* __restrict__ batch,
                           float* __restrict__ c, int n) {
    int i = blockIdx.x * blockDim.x + threadIdx.x;
    if (i < n) atomicAdd(&c[batch[i]], 1.0f);
}

__global__ void pool_finalize(float* __restrict__ accum,
                              const float* __restrict__ s,
                              const float* __restrict__ c) {
    int i = blockIdx.x * blockDim.x + threadIdx.x;  // B*H threads
    accum[i] += s[i] / fmaxf(c[i >> 5], 1.0f);
}

// ---------------- final head: logits + softmax over axis 0 ----------------
__global__ void final_head(const float* __restrict__ accum,
                           const float* __restrict__ Wl,
                           const float* __restrict__ bl,
                           float* __restrict__ out) {
    __shared__ float lg[kOUT * kB];       // column-major: lg[o*256 + r]
    __shared__ float cmax[kOUT], csum[kOUT];
    const int r = threadIdx.x;            // one graph per thread (256)

    float a[kH];
    #pragma unroll
    for (int k = 0; k < kH; ++k) a[k] = accum[(size_t)r * kH + k];

    #pragma unroll
    for (int o = 0; o < kOUT; ++o) {
        float s = bl[o];
        #pragma unroll
        for (int k = 0; k < kH; ++k) s += a[k] * Wl[k * kOUT + o];
        lg[o * kB + r] = s;
    }
    __syncthreads();

    if (r < kOUT) {
        float m = -INFINITY;
        for (int i = 0; i < kB; ++i) m = fmaxf(m, lg[r * kB + i]);
        float s = 0.0f;
        for (int i = 0; i < kB; ++i) s += expf(lg[r * kB + i] - m);
        cmax[r] = m; csum[r] = s;
    }
    __syncthreads();

    #pragma unroll
    for (int o = 0; o < kOUT; ++o)
        out[(size_t)r * kOUT + o] = expf(lg[o * kB + r] - cmax[o]) / csum[o];
}

// ---------------- host orchestration ----------------
static inline void launch_fill(float* p, long long n, hipStream_t s) {
    fill0<<<(unsigned)((n + 255) / 256), 256, 0, s>>>(p, n);
}

extern "C" void kernel_launch(void* const* d_in, const int* in_sizes, int n_in,
                              void* d_out, int out_size, void* d_ws, size_t ws_size,
                              hipStream_t stream) {
    (void)in_sizes; (void)n_in; (void)out_size; (void)ws_size;

    const float* x[3]   = {(const float*)d_in[0], (const float*)d_in[1], (const float*)d_in[2]};
    const int*   row[3]; const int* colp[3]; const float* val[3]; const int* bat[3];
    for (int i = 0; i < 3; ++i) {
        row[i]  = (const int*)  d_in[3 + 4 * i];
        colp[i] = (const int*)  d_in[4 + 4 * i];
        val[i]  = (const float*)d_in[5 + 4 * i];
        bat[i]  = (const int*)  d_in[6 + 4 * i];
    }
    const float* Wm[3][3]; const float* bm[3][3];
    for (int br = 0; br < 3; ++br)
        for (int l = 0; l < 3; ++l) {
            Wm[br][l] = (const float*)d_in[15 + 6 * br + 2 * l];
            bm[br][l] = (const float*)d_in[16 + 6 * br + 2 * l];
        }
    const float* Wl = (const float*)d_in[33];
    const float* bl = (const float*)d_in[34];

    // workspace layout (floats): two 8M ping-pong buffers + pooling scratch
    float* ws     = (float*)d_ws;
    float* zbuf   = ws;                                  // 8,388,608
    float* hbuf   = ws + (size_t)8388608;                // 8,388,608
    float* pool_s = ws + (size_t)16777216;               // 8192
    float* pool_c = pool_s + (size_t)kB * kH;            // 256
    float* accum  = pool_c + kB;                         // 8192

    const int       nArr[3]   = {kN0, kN1, kN2};
    const int       finArr[3] = {kF1, kF2, kF3};

    launch_fill(accum, (long long)kB * kH, stream);

    for (int br = 0; br < 3; ++br) {
        const int n   = nArr[br];
        const int fin = finArr[br];
        const long long nnz = (long long)n * kDEG;
        const unsigned sgrid = (unsigned)(nnz / 256);          // 32 edges/wave, 8 waves/block
        const unsigned ggrid = (unsigned)((n / 16) * 2 / 8);   // 16x16 tiles, 8 waves/block

        // ---- layer 1: z = L @ leaky(x);  h = z @ W1 + b1 ----
        launch_fill(zbuf, (long long)n * fin, stream);
        if (fin == 64) {
            spmm_kernel<64, 1><<<sgrid, 256, 0, stream>>>(row[br], colp[br], val[br], x[br], zbuf, nnz);
            gemm_wmma<64><<<ggrid, 256, 0, stream>>>(zbuf, Wm[br][0], bm[br][0], hbuf);
        } else {
            spmm_kernel<32, 1><<<sgrid, 256, 0, stream>>>(row[br], colp[br], val[br], x[br], zbuf, nnz);
            gemm_wmma<32><<<ggrid, 256, 0, stream>>>(zbuf, Wm[br][0], bm[br][0], hbuf);
        }

        // ---- layer 2: z = L @ leaky(h);  h = z @ W2 + b2 ----
        launch_fill(zbuf, (long long)n * kH, stream);
        spmm_kernel<32, 1><<<sgrid, 256, 0, stream>>>(row[br], colp[br], val[br], hbuf, zbuf, nnz);
        gemm_wmma<32><<<ggrid, 256, 0, stream>>>(zbuf, Wm[br][1], bm[br][1], hbuf);

        // ---- layer 3: z = L @ h;  h = z @ W3 + b3 (no activation) ----
        launch_fill(zbuf, (long long)n * kH, stream);
        spmm_kernel<32, 0><<<sgrid, 256, 0, stream>>>(row[br], colp[br], val[br], hbuf, zbuf, nnz);
        gemm_wmma<32><<<ggrid, 256, 0, stream>>>(zbuf, Wm[br][2], bm[br][2], hbuf);

        // ---- mean pool into accumulator ----
        launch_fill(pool_s, (long long)kB * kH, stream);
        launch_fill(pool_c, kB, stream);
        pool_sum<<<(unsigned)(((long long)n * kH) / 256), 256, 0, stream>>>(hbuf, bat[br], pool_s);
        pool_count<<<(unsigned)(n / 256), 256, 0, stream>>>(bat[br], pool_c, n);
        pool_finalize<<<(kB * kH) / 256, 256, 0, stream>>>(accum, pool_s, pool_c);
    }

    final_head<<<1, 256, 0, stream>>>(accum, Wl, bl, (float*)d_out);
}